// _QLSTMCell_65481071398004
// MI455X (gfx1250) — compile-verified
//
#include <hip/hip_runtime.h>

#define N_WIRES 12
#define BATCH   2048
#define TPB     256

// Closed form of the reference circuit:
//   out[b,0]   = prod_{u=1..11} cos(x[b,u]+theta[u])
//   out[b,w>0] = prod_{u=0..w } cos(x[b,u]+theta[u])
// Derivation: RX gates on |0..0> keep a product state => <Z_u> = cos(x_u+theta_u);
// the CNOT ring is GF(2)-linear, measured bit 0 = XOR(bits 1..11),
// bit w = XOR(bits 0..w) => expectations factor into the products above.
__global__ __launch_bounds__(TPB) void qexpval_kernel(
    const float* __restrict__ x,
    const float* __restrict__ theta,
    float* __restrict__ out)
{
    // Grid exactly covers BATCH (8 blocks x 256 threads) -> no bounds check.
    const int b = blockIdx.x * TPB + threadIdx.x;

    // Uniform theta: 12 floats, hot in WGP$ after the first wave.
    float th[N_WIRES];
#pragma unroll
    for (int u = 0; u < N_WIRES; ++u) th[u] = theta[u];

    // Row load: 48 bytes, 16B aligned -> 3x global_load_b128.
    const float4* __restrict__ xv = reinterpret_cast<const float4*>(x + b * N_WIRES);
    const float4 a0 = xv[0];
    const float4 a1 = xv[1];
    const float4 a2 = xv[2];

    const float xs[N_WIRES] = { a0.x, a0.y, a0.z, a0.w,
                                a1.x, a1.y, a1.z, a1.w,
                                a2.x, a2.y, a2.z, a2.w };

    // Native cosine: v_mul_f32 (x * 1/2pi) + v_cos_f32 (TRANS op, co-executes
    // with VALU on CDNA5). Args are ~N(0,2) -> native accuracy ~1e-6 abs.
    float c[N_WIRES];
#pragma unroll
    for (int u = 0; u < N_WIRES; ++u) c[u] = __cosf(xs[u] + th[u]);

    float res[N_WIRES];
    // Prefix products for wires 1..11 (include wire 0's factor).
    float pre = c[0];
#pragma unroll
    for (int w = 1; w < N_WIRES; ++w) { pre *= c[w]; res[w] = pre; }
    // Wire 0: product of c[1..11] (computed directly; no division so c[0]==0 is safe).
    float q = c[1];
#pragma unroll
    for (int u = 2; u < N_WIRES; ++u) q *= c[u];
    res[0] = q;

    // Row store: 3x global_store_b128.
    float4* __restrict__ ov = reinterpret_cast<float4*>(out + b * N_WIRES);
    ov[0] = make_float4(res[0], res[1], res[2],  res[3]);
    ov[1] = make_float4(res[4], res[5], res[6],  res[7]);
    ov[2] = make_float4(res[8], res[9], res[10], res[11]);
}

extern "C" void kernel_launch(void* const* d_in, const int* in_sizes, int n_in,
                              void* d_out, int out_size, void* d_ws, size_t ws_size,
                              hipStream_t stream)
{
    const float* x     = (const float*)d_in[0];   // (2048, 12) fp32
    const float* theta = (const float*)d_in[1];   // (12,)      fp32
    float*       out   = (float*)d_out;           // (2048, 12) fp32

    (void)in_sizes; (void)n_in; (void)out_size; (void)d_ws; (void)ws_size;

    qexpval_kernel<<<dim3(BATCH / TPB), dim3(TPB), 0, stream>>>(x, theta, out);
}